// CRF_90744069030115
// MI455X (gfx1250) — compile-verified
//
#include <hip/hip_runtime.h>
#include <hip/hip_bf16.h>

// Viterbi CRF decode for MI455X (gfx1250).
// S=4096, B=128, T=48. One 64-thread block (2 wave32) per batch element.
// - transitions column held in 48 VGPRs per thread (constant over the scan)
// - previous scores broadcast from LDS as 12 x ds_load_b128 per step
// - 4-way split max/argmax chains (index-order merge preserves first-max)
// - emissions/mask double-buffered via global_load_async_to_lds_b128/b32
// - u8 backpointers to d_ws; backward pointer-chase over async-loaded chunks

#define SEQ   4096
#define BATCH 128
#define NT    48
#define ROWB  (NT * 4)        // 192 bytes per (t,b) emission row (12 x 16B)
#define BPROW (BATCH * NT)    // 6144 bytes per backpointer timestep

// ---------------- CDNA5 async global->LDS path (probe-verified) ------------
#if defined(__has_builtin)
#  if __has_builtin(__builtin_amdgcn_global_load_async_to_lds_b128) && \
      __has_builtin(__builtin_amdgcn_global_load_async_to_lds_b32)
#    define HAS_ASYNC 1
#  endif
#endif
#ifndef HAS_ASYNC
#  define HAS_ASYNC 0
#endif

typedef int v4i __attribute__((ext_vector_type(4)));

__device__ __forceinline__ void async_b128(const void* g, void* l) {
#if HAS_ASYNC
  __builtin_amdgcn_global_load_async_to_lds_b128(
      (__attribute__((address_space(1))) v4i*)g,
      (__attribute__((address_space(3))) v4i*)l, 0, 0);
#else
  *(int4*)l = *(const int4*)g;
#endif
}
__device__ __forceinline__ void async_b32(const void* g, void* l) {
#if HAS_ASYNC
  __builtin_amdgcn_global_load_async_to_lds_b32(
      (__attribute__((address_space(1))) int*)g,
      (__attribute__((address_space(3))) int*)l, 0, 0);
#else
  *(int*)l = *(const int*)g;
#endif
}

#if HAS_ASYNC
#  if __has_builtin(__builtin_amdgcn_s_wait_asynccnt)
#    define ASYNC_WAIT(n) __builtin_amdgcn_s_wait_asynccnt(n)
#  else
#    define ASYNC_WAIT(n) asm volatile("s_wait_asynccnt %0" ::"i"(n) : "memory")
#  endif
#else
#  define ASYNC_WAIT(n) do { } while (0)
#endif

// ---------------- LDS layout ----------------------------------------------
struct __align__(16) Smem {
  float score[2][NT];               // ping-pong Viterbi scores (16B-aligned rows)
  union {
    struct {                        // forward phase
      float em[2][64 * NT];         // 2 x (64 steps x 48 emissions)
      int   msk[2][64];             // 2 x (64 steps of mask)
    } f;
    struct {                        // backtrace phase
      unsigned char bp[2][256 * NT];// 2 x (256 steps x 48 backpointers)
      int stage[256];               // decoded tags staging
    } bt;
  } u;
};

// Emission/mask chunk c (steps t = 1+64c ..) into buffer pb.
// Exactly 12 b128 + 1 b32 per wave (address clamping keeps EXEC full so
// ASYNCcnt advances by a fixed +13 per wave per chunk).
__device__ __forceinline__ void issue_em(const float* em, const int* mask,
                                         int b, int c, int pb, Smem& S) {
  const int tid = threadIdx.x;
  const int tfirst = 1 + (c << 6);
  const char* eb = (const char*)em;
  char* lb = (char*)&S.u.f.em[pb][0];
#pragma unroll
  for (int it = 0; it < 12; ++it) {
    int lin = tid + (it << 6);
    int r = lin / 12, d = lin - r * 12;          // r: step-in-chunk, d: 16B piece
    int t = tfirst + r; if (t > SEQ - 1) t = SEQ - 1;
    async_b128(eb + ((size_t)t * BATCH + b) * ROWB + d * 16,
               lb + r * ROWB + d * 16);
  }
  int r = tid;
  int t = tfirst + r; if (t > SEQ - 1) t = SEQ - 1;
  async_b32((const char*)mask + ((size_t)t * BATCH + b) * 4,
            (char*)&S.u.f.msk[pb][0] + r * 4);
}

// Backpointer chunk k (t = SEQ-1-256k downward) into buffer pb; 12 b128/wave.
__device__ __forceinline__ void issue_bp(const unsigned char* bp, int b, int k,
                                         int pb, Smem& S) {
  const int tid = threadIdx.x;
  int tHi = SEQ - 1 - (k << 8);
  int tLo = tHi - 255; if (tLo < 1) tLo = 1;
  const char* gb = (const char*)bp + (size_t)b * NT; // b*48 is 16B aligned
  char* lb = (char*)&S.u.bt.bp[pb][0];
#pragma unroll
  for (int it = 0; it < 12; ++it) {
    int lin = tid + (it << 6);
    int r = lin / 3, d = lin - r * 3;            // 3 x b128 per 48B row
    int t = tLo + r; if (t > tHi) t = tHi;
    async_b128(gb + (size_t)(t - 1) * BPROW + d * 16,
               lb + r * NT + d * 16);
  }
}

// strict > keeps the FIRST maximal index (matches jnp.argmax)
#define UPD(bb, aa, val, idx)                \
  do {                                       \
    float v_ = (val);                        \
    bool g_ = (v_ > (bb));                   \
    (bb) = g_ ? v_ : (bb);                   \
    (aa) = g_ ? (idx) : (aa);                \
  } while (0)

__global__ __launch_bounds__(64)
void crf_viterbi_kernel(const float* __restrict__ em,
                        const int*   __restrict__ mask,
                        const float* __restrict__ startt,
                        const float* __restrict__ endt,
                        const float* __restrict__ trans,
                        int* __restrict__ out,
                        unsigned char* __restrict__ bp) {
  __shared__ Smem S;
  const int tid = threadIdx.x;
  const int b   = blockIdx.x;

  // ---- transitions column for this thread's tag -> 48 VGPRs (scan-constant)
  float rt[NT];
  if (tid < NT) {
#pragma unroll
    for (int i = 0; i < NT; ++i) rt[i] = trans[i * NT + tid];  // trans[i][tid]
    S.score[0][tid] = startt[tid] + em[(size_t)b * NT + tid];  // t=0 row
  }

  issue_em(em, mask, b, 0, 0, S);                // prefetch chunk 0 (t=1..64)
  __syncthreads();

  // ---- forward Viterbi scan, double-buffered 64-step chunks ---------------
  int t = 1;
  const int NCH = 64;
  for (int c = 0; c < NCH; ++c) {
    if (c + 1 < NCH) { issue_em(em, mask, b, c + 1, (c + 1) & 1, S); ASYNC_WAIT(13); }
    else             { ASYNC_WAIT(0); }
    __syncthreads();                             // chunk c resident for both waves
    int cnt = SEQ - 1 - (c << 6); if (cnt > 64) cnt = 64;
    const int pb = c & 1;
    for (int r = 0; r < cnt; ++r, ++t) {
      if (tid < NT) {
        const int p = (t - 1) & 1;
        // broadcast previous scores: 12 x ds_load_b128 (uniform address)
        float s[NT];
#pragma unroll
        for (int i4 = 0; i4 < 12; ++i4) {
          float4 v = ((const float4*)&S.score[p][0])[i4];
          s[i4 * 4 + 0] = v.x; s[i4 * 4 + 1] = v.y;
          s[i4 * 4 + 2] = v.z; s[i4 * 4 + 3] = v.w;
        }
        // 4 independent max/argmax chains over contiguous quarters
        float b0 = -__builtin_inff(), b1 = b0, b2 = b0, b3 = b0;
        int a0 = 0, a1 = 12, a2 = 24, a3 = 36;
#pragma unroll
        for (int q = 0; q < 12; ++q) {
          UPD(b0, a0, s[q]      + rt[q],      q);
          UPD(b1, a1, s[12 + q] + rt[12 + q], 12 + q);
          UPD(b2, a2, s[24 + q] + rt[24 + q], 24 + q);
          UPD(b3, a3, s[36 + q] + rt[36 + q], 36 + q);
        }
        // merge in ascending index order (strict > => first-max wins)
        if (b1 > b0) { b0 = b1; a0 = a1; }
        if (b2 > b0) { b0 = b2; a0 = a2; }
        if (b3 > b0) { b0 = b3; a0 = a3; }

        float ns = b0 + S.u.f.em[pb][r * NT + tid];
        int m = S.u.f.msk[pb][r];
        S.score[t & 1][tid] = m ? ns : s[tid];
        bp[(size_t)(t - 1) * BPROW + (size_t)b * NT + tid] = (unsigned char)a0;
      }
      __syncthreads();                           // single barrier per step
    }
  }

  // make backpointer stores visible before the async engine reads them back
  asm volatile("s_wait_storecnt 0" ::: "memory");
  __syncthreads();

  // ---- final argmax over score + end_transitions --------------------------
  if (tid < NT)
    S.score[0][tid] = endt[tid] + S.score[(SEQ - 1) & 1][tid];
  __syncthreads();
  int tag = 0;                                   // live only in thread 0
  if (tid == 0) {
    float best = -__builtin_inff();
    for (int j = 0; j < NT; ++j) {
      float v = S.score[0][j];
      if (v > best) { best = v; tag = j; }
    }
    out[(size_t)b * SEQ + (SEQ - 1)] = tag;
  }

  // ---- backward pointer-chase over async-loaded 256-step chunks -----------
  issue_bp(bp, b, 0, 0, S);
  const int NB = 16;
  for (int k = 0; k < NB; ++k) {
    if (k + 1 < NB) { issue_bp(bp, b, k + 1, (k + 1) & 1, S); ASYNC_WAIT(12); }
    else            { ASYNC_WAIT(0); }
    __syncthreads();                             // chunk k resident
    int tHi = SEQ - 1 - (k << 8);
    int tLo = tHi - 255; if (tLo < 1) tLo = 1;
    const int pb = k & 1;
    if (tid == 0) {                              // serial walk at LDS latency
      int tg = tag;
      for (int tt = tHi; tt >= tLo; --tt) {
        tg = S.u.bt.bp[pb][(tt - tLo) * NT + tg];
        S.u.bt.stage[tt - tLo] = tg;             // tag at time tt-1
      }
      tag = tg;
    }
    __syncthreads();
    int n = tHi - tLo + 1;
    for (int i = tid; i < n; i += 64)
      out[(size_t)b * SEQ + (tLo - 1 + i)] = S.u.bt.stage[i];
    __syncthreads();                             // protect stage reuse
  }
}

extern "C" void kernel_launch(void* const* d_in, const int* in_sizes, int n_in,
                              void* d_out, int out_size, void* d_ws, size_t ws_size,
                              hipStream_t stream) {
  (void)in_sizes; (void)n_in; (void)out_size; (void)ws_size;
  const float* em    = (const float*)d_in[0];   // [4096,128,48] f32
  const int*   mask  = (const int*)d_in[1];     // [4096,128] i32
  const float* st    = (const float*)d_in[2];   // [48]
  const float* en    = (const float*)d_in[3];   // [48]
  const float* tr    = (const float*)d_in[4];   // [48,48]
  int* out = (int*)d_out;                       // [128,4096] int32 tags
  unsigned char* bp = (unsigned char*)d_ws;     // 4095*128*48 bytes backpointers

  hipLaunchKernelGGL(crf_viterbi_kernel, dim3(BATCH), dim3(64), 0, stream,
                     em, mask, st, en, tr, out, bp);
}